// RelPN_55619826483618
// MI455X (gfx1250) — compile-verified
//
#include <hip/hip_runtime.h>

// ---------------------------------------------------------------------------
// RelPN proposal-pair matching, MI455X (gfx1250) version.
// quality [1024 x 65280] f32 is store-bandwidth bound (~267 MB / 23.3 TB/s).
// Tiles of quality are exact 0.5-scaled copies of iou sub-blocks plus a
// broadcast constant -> built with V_WMMA_F32_16X16X4_F32 (identity x iou + C),
// streamed out with non-temporal stores. Max/argmax is separable so the
// 1024-way reduction collapses to two 32-way reductions computed once.
// ---------------------------------------------------------------------------

#define N_PROP 256
#define N_TGT  32
#define NN1    (N_PROP * (N_PROP - 1))   // 65280 off-diagonal proposal pairs
#define MM     (N_TGT * N_TGT)           // 1024 GT pairs

// d_out layout (floats, reference tuple order, concatenated flat)
#define QUAL_SZ  ((size_t)MM * (size_t)NN1)          // 66,846,720
#define OFF_MIDX (QUAL_SZ)
#define OFF_LAB  (OFF_MIDX + (size_t)NN1)
#define OFF_PBP  (OFF_LAB  + (size_t)NN1)
#define OFF_PIP  (OFF_PBP  + (size_t)NN1 * 8)
#define OFF_TBP  (OFF_PIP  + (size_t)NN1 * 2)

typedef float v2f __attribute__((ext_vector_type(2)));
typedef float v8f __attribute__((ext_vector_type(8)));

// --------------------------------------------------------------------------
// Kernel A: iou[32,256] once, plus per-proposal column max / first-argmax.
// --------------------------------------------------------------------------
__global__ void k_iou(const float* __restrict__ prop, const float* __restrict__ tgt,
                      float* __restrict__ iou, float* __restrict__ maxv,
                      int* __restrict__ argm) {
    const int i = threadIdx.x;                 // proposal index, 256 threads
    const float px0 = prop[i * 4 + 0], py0 = prop[i * 4 + 1];
    const float px1 = prop[i * 4 + 2], py1 = prop[i * 4 + 3];
    const float areaP = (px1 - px0 + 1.0f) * (py1 - py0 + 1.0f);
    float best = -1.0f; int bestA = 0;
    for (int a = 0; a < N_TGT; ++a) {
        const float tx0 = tgt[a * 4 + 0], ty0 = tgt[a * 4 + 1];
        const float tx1 = tgt[a * 4 + 2], ty1 = tgt[a * 4 + 3];
        const float areaT = (tx1 - tx0 + 1.0f) * (ty1 - ty0 + 1.0f);
        float w = fminf(tx1, px1) - fmaxf(tx0, px0) + 1.0f;
        float h = fminf(ty1, py1) - fmaxf(ty0, py0) + 1.0f;
        w = fmaxf(w, 0.0f); h = fmaxf(h, 0.0f);
        const float inter = w * h;
        const float v = inter / (areaT + areaP - inter);
        iou[a * N_PROP + i] = v;
        if (v > best) { best = v; bestA = a; }   // strict > keeps FIRST argmax
    }
    maxv[i] = best;
    argm[i] = bestA;
}

// --------------------------------------------------------------------------
// Kernel B: quality via WMMA.  One wave owns (a, 16-row b-half) for subject
// proposal i = blockIdx.x and loops all 16 column tiles (j-blocks).
// Tile:  D[r,c] = 0.5*iou[a,i] + 0.5*iou[b0+r, j0+c]
//      = sum_{t=0..3} (0.5*I4_t) x iou_block_t  +  C(const)     (bit exact)
// --------------------------------------------------------------------------
__global__ void __launch_bounds__(256) k_quality(const float* __restrict__ iou,
                                                 float* __restrict__ out) {
    const int i    = blockIdx.x;            // subject proposal
    const int wave = threadIdx.x >> 5;      // wave32: 8 waves per block
    const int lane = threadIdx.x & 31;
    const int t    = blockIdx.y * 8 + wave; // 0..63 -> (a, b-half)
    const int a    = t & 31;
    const int b0   = (t >> 5) * 16;         // 0 or 16
    const int n    = lane & 15;             // A-row (M) for A operand; N for B/D
    const int hi   = lane >> 4;             // lane half selects K pair / D row+8
    const int k0   = hi ? 2 : 0;

    __builtin_prefetch(iou, 0, 3);          // global_prefetch_b8: keep iou hot

    const float c0 = 0.5f * iou[a * N_PROP + i];   // broadcast tile constant

    for (int jb = 0; jb < 16; ++jb) {
        const int j0 = jb * 16;
        v8f acc;
        #pragma unroll
        for (int v = 0; v < 8; ++v) acc[v] = c0;

        #pragma unroll
        for (int step = 0; step < 4; ++step) {
            // A = 0.5-scaled partial identity rows [4*step, 4*step+4)
            v2f A, B;
            A.x = (n == 4 * step + k0    ) ? 0.5f : 0.0f;
            A.y = (n == 4 * step + k0 + 1) ? 0.5f : 0.0f;
            // B[k,c] = iou[b0 + 4*step + k, j0 + c]
            const int rB = b0 + 4 * step;
            B.x = iou[(rB + k0    ) * N_PROP + (j0 + n)];
            B.y = iou[(rB + k0 + 1) * N_PROP + (j0 + n)];
            acc = __builtin_amdgcn_wmma_f32_16x16x4_f32(
                false, A, false, B, (short)0, acc, false, false);
        }

        // D layout: lane holds col c = n, rows m = v + hi*8 -> b = b0 + m.
        const int j = j0 + n;
        if (j != i) {                                    // drop diagonal pair
            const long long col = (long long)i * (N_PROP - 1) + j - (j > i);
            #pragma unroll
            for (int v = 0; v < 8; ++v) {
                const int b = b0 + v + hi * 8;
                const long long p = (long long)a * N_TGT + b;
                __builtin_nontemporal_store(acc[v],
                    &out[p * (long long)NN1 + col]);     // streaming store
            }
        }
    }
}

// --------------------------------------------------------------------------
// Kernel C: matched_idxs / labels via separable max, plus box & index pairs.
// --------------------------------------------------------------------------
__global__ void k_small(const float* __restrict__ prop, const float* __restrict__ tgt,
                        const int* __restrict__ pairLabels,
                        const float* __restrict__ maxv, const int* __restrict__ argm,
                        float* __restrict__ out) {
    const int i = blockIdx.x;
    const int j = threadIdx.x;

    if (j != i) {
        const long long col = (long long)i * (N_PROP - 1) + j - (j > i);
        const float mv = 0.5f * (maxv[i] + maxv[j]);
        const int match = argm[i] * N_TGT + argm[j];     // first flat argmax
        const int midx  = (mv < 0.3f) ? -1 : ((mv < 0.5f) ? -2 : match);
        float lab;
        if      (midx == -1) lab = 0.0f;                 // background
        else if (midx == -2) lab = -1.0f;                // between thresholds
        else                 lab = (float)pairLabels[match];
        out[OFF_MIDX + col] = (float)midx;
        out[OFF_LAB  + col] = lab;
        #pragma unroll
        for (int k = 0; k < 4; ++k) {
            out[OFF_PBP + col * 8 + k]     = prop[i * 4 + k];
            out[OFF_PBP + col * 8 + 4 + k] = prop[j * 4 + k];
        }
        out[OFF_PIP + col * 2]     = (float)i;
        out[OFF_PIP + col * 2 + 1] = (float)j;
    }

    if (i < 4) {                                         // 4*256 = 1024 GT pairs
        const int p = i * 256 + j;
        const int a = p >> 5, b = p & 31;
        #pragma unroll
        for (int k = 0; k < 4; ++k) {
            out[OFF_TBP + (long long)p * 8 + k]     = tgt[a * 4 + k];
            out[OFF_TBP + (long long)p * 8 + 4 + k] = tgt[b * 4 + k];
        }
    }
}

// --------------------------------------------------------------------------
extern "C" void kernel_launch(void* const* d_in, const int* in_sizes, int n_in,
                              void* d_out, int out_size, void* d_ws, size_t ws_size,
                              hipStream_t stream) {
    const float* prop       = (const float*)d_in[0];   // [256,4]
    const float* tgt        = (const float*)d_in[1];   // [32,4]
    const int*   pairLabels = (const int*)d_in[2];     // [1024]
    float* out = (float*)d_out;

    float* iou  = (float*)d_ws;                        // 32*256 f32
    float* maxv = iou + N_TGT * N_PROP;                // 256 f32
    int*   argm = (int*)(maxv + N_PROP);               // 256 i32

    k_iou    <<<1, N_PROP, 0, stream>>>(prop, tgt, iou, maxv, argm);
    k_quality<<<dim3(N_PROP, 8), 256, 0, stream>>>(iou, out);
    k_small  <<<N_PROP, N_PROP, 0, stream>>>(prop, tgt, pairLabels, maxv, argm, out);
}